// MiccaiSurvLoss_46600395162160
// MI455X (gfx1250) — compile-verified
//
#include <hip/hip_runtime.h>
#include <hip/hip_bf16.h>
#include <stdint.h>

// ---------------- problem geometry ----------------
#define BROWS   1048576u
#define KDIM    32
#define CHUNK   256                 // rows per block in K1/K3
#define NB      (BROWS / CHUNK)     // 4096 blocks
#define WPB     8                   // waves per 256-thread block (wave32)
#define RPW     (CHUNK / WPB)       // 32 rows per wave
#define SUBR    16                  // rows per TDM subtile (16*32*4B = 2KB)
#define NSUB    (RPW / SUBR)        // 2 subtiles per wave (double buffer)
#define LOGEPS  (-16.118095650958319f)   // log(1e-7)

typedef unsigned int u32;
typedef __attribute__((ext_vector_type(4))) unsigned int uv4;
typedef __attribute__((ext_vector_type(4))) int          iv4;
typedef __attribute__((ext_vector_type(8))) int          iv8;

#if __has_builtin(__builtin_amdgcn_tensor_load_to_lds) && __has_builtin(__builtin_amdgcn_s_wait_tensorcnt)
#define USE_TDM 1
#else
#define USE_TDM 0
#endif

__device__ __forceinline__ float waveSum32(float v) {
#pragma unroll
  for (int m = 16; m >= 1; m >>= 1) v += __shfl_xor(v, m, 32);
  return v;
}

// ---------------- K1: per-block max of a = (c==1 ? y : -1) ----------------
__global__ __launch_bounds__(CHUNK) void k1_blockmax(const int* __restrict__ y,
                                                     const int* __restrict__ c,
                                                     int* __restrict__ blockmax) {
  __shared__ int red[CHUNK];
  const int i = threadIdx.x;
  const u32 r = blockIdx.x * (u32)CHUNK + (u32)i;
  int a = (c[r] == 1) ? y[r] : -1;
  red[i] = a;
  __syncthreads();
#pragma unroll
  for (int off = CHUNK / 2; off > 0; off >>= 1) {
    if (i < off) red[i] = max(red[i], red[i + off]);
    __syncthreads();
  }
  if (i == 0) blockmax[blockIdx.x] = red[0];
}

// ------- K2: exclusive forward-prefix-max and backward-suffix-max over NB=4096 -------
__global__ __launch_bounds__(1024) void k2_scan(const int* __restrict__ bm,
                                                int* __restrict__ fwd,
                                                int* __restrict__ bwd) {
  __shared__ int sc[1024];
  const int t = threadIdx.x;
  const int m0 = bm[t * 4 + 0], m1 = bm[t * 4 + 1], m2 = bm[t * 4 + 2], m3 = bm[t * 4 + 3];
  const int agg = max(max(m0, m1), max(m2, m3));

  // forward inclusive Hillis-Steele over 1024 thread aggregates
  int v = agg;
  sc[t] = v;
  __syncthreads();
  for (int off = 1; off < 1024; off <<= 1) {
    int o = (t >= off) ? sc[t - off] : -1;
    __syncthreads();
    v = max(v, o);
    sc[t] = v;
    __syncthreads();
  }
  int epre = (t > 0) ? sc[t - 1] : -1;   // exclusive prefix for this thread's group
  fwd[t * 4 + 0] = epre;
  int p = max(epre, m0); fwd[t * 4 + 1] = p;
  p = max(p, m1);        fwd[t * 4 + 2] = p;
  p = max(p, m2);        fwd[t * 4 + 3] = p;
  __syncthreads();

  // backward inclusive scan
  v = agg;
  sc[t] = v;
  __syncthreads();
  for (int off = 1; off < 1024; off <<= 1) {
    int o = (t + off < 1024) ? sc[t + off] : -1;
    __syncthreads();
    v = max(v, o);
    sc[t] = v;
    __syncthreads();
  }
  int esuf = (t < 1023) ? sc[t + 1] : -1;  // exclusive suffix for this thread's group
  bwd[t * 4 + 3] = esuf;
  int s = max(esuf, m3); bwd[t * 4 + 2] = s;
  s = max(s, m2);        bwd[t * 4 + 1] = s;
  s = max(s, m1);        bwd[t * 4 + 0] = s;
}

// ---------------- K3: main loss kernel ----------------
__global__ __launch_bounds__(CHUNK) void k3_main(const float* __restrict__ h,
                                                 const int* __restrict__ y,
                                                 const int* __restrict__ c,
                                                 const int* __restrict__ fwd,
                                                 const int* __restrict__ bwd,
                                                 float* __restrict__ partial) {
  __shared__ float hstage[WPB * NSUB * SUBR * KDIM];  // 32 KB: per-wave double-buffered tiles
  __shared__ int yL[CHUNK], cL[CHUNK], pmaxL[CHUNK], smaxL[CHUNK], tmpL[CHUNK];
  __shared__ float wacc[WPB];

  const int i = threadIdx.x;
  const u32 b = blockIdx.x;
  const u32 row0 = b * (u32)CHUNK;
  const int w = i >> 5;
  const int lane = i & 31;

  // ---- phase 0: chunk-local prefix/suffix max of a, seeded by cross-block scans ----
  const int yv = y[row0 + (u32)i];
  const int cv = c[row0 + (u32)i];
  const int a = (cv == 1) ? yv : -1;
  yL[i] = yv;
  cL[i] = cv;

  int v = a;
  tmpL[i] = v;
  __syncthreads();
  for (int off = 1; off < CHUNK; off <<= 1) {
    int o = (i >= off) ? tmpL[i - off] : -1;
    __syncthreads();
    v = max(v, o);
    tmpL[i] = v;
    __syncthreads();
  }
  pmaxL[i] = max(v, fwd[b]);      // inclusive global prefix max
  __syncthreads();

  v = a;
  tmpL[i] = v;
  __syncthreads();
  for (int off = 1; off < CHUNK; off <<= 1) {
    int o = (i + off < CHUNK) ? tmpL[i + off] : -1;
    __syncthreads();
    v = max(v, o);
    tmpL[i] = v;
    __syncthreads();
  }
  smaxL[i] = max((i + 1 < CHUNK) ? tmpL[i + 1] : -1, bwd[b]);  // exclusive global suffix max
  __syncthreads();

  // ---- phase 1: TDM-stage this wave's h rows into LDS (2 x 2KB, double buffered) ----
  const u32 ldsBase = (u32)(size_t)&hstage[(w * NSUB) * SUBR * KDIM];
  const uint64_t gbase = (uint64_t)(size_t)h + (uint64_t)(row0 + (u32)(w * RPW)) * (KDIM * 4);

#if USE_TDM
#pragma unroll
  for (int s = 0; s < NSUB; ++s) {
    const uint64_t ga = gbase + (uint64_t)s * (SUBR * KDIM * 4);
    u32 lo = (u32)__builtin_amdgcn_readfirstlane((int)(u32)(ga & 0xffffffffull));
    u32 hi = (u32)__builtin_amdgcn_readfirstlane((int)(u32)(ga >> 32));
    u32 la = (u32)__builtin_amdgcn_readfirstlane((int)(ldsBase + (u32)(s * SUBR * KDIM * 4)));
    uv4 g0;
    g0[0] = 1u;                                   // count=1, user descriptor
    g0[1] = la;                                   // lds_addr (bytes)
    g0[2] = lo;                                   // global_addr[31:0]
    g0[3] = (hi & 0x01ffffffu) | (2u << 30);      // global_addr[56:32] | type=2
    iv8 g1;
    g1[0] = 0x00020000;                           // data_size = 4B; no multicast/pad/iterate
    g1[1] = (int)(512u << 16);                    // tensor_dim0 = 512 elements (one subtile)
    g1[2] = (int)(1u << 16);                      // tensor_dim1 = 1
    g1[3] = (int)(512u << 16);                    // tile_dim0 = 512
    g1[4] = 1;                                    // tile_dim1 = 1
    g1[5] = 512;                                  // tensor_dim0_stride = 512
    g1[6] = 0;
    g1[7] = 0;
    iv4 g2; g2[0] = 1; g2[1] = 1; g2[2] = 0; g2[3] = 0;
    iv4 g3; g3[0] = 0; g3[1] = 0; g3[2] = 0; g3[3] = 0;
    iv8 g4;                                        // 6-arg toolchain: extra 256-bit group (zero)
    g4[0] = 0; g4[1] = 0; g4[2] = 0; g4[3] = 0;
    g4[4] = 0; g4[5] = 0; g4[6] = 0; g4[7] = 0;
    __builtin_amdgcn_tensor_load_to_lds(g0, g1, g2, g3, g4, 0);
  }
#else
  // fallback: cooperative load (wave writes only its own LDS region)
#pragma unroll
  for (int s = 0; s < NSUB; ++s)
    for (int q = lane; q < SUBR * KDIM; q += 32)
      hstage[(w * NSUB + s) * SUBR * KDIM + q] =
          h[(size_t)(row0 + (u32)(w * RPW + s * SUBR)) * KDIM + (u32)q];
#endif

  // ---- phase 2: one wave per row; lane k == feature k ----
  float acc = 0.0f;
#pragma unroll
  for (int s = 0; s < NSUB; ++s) {
#if USE_TDM
    if (s == 0) __builtin_amdgcn_s_wait_tensorcnt(1);  // first tile ready, 2nd in flight
    else        __builtin_amdgcn_s_wait_tensorcnt(0);  // all tiles ready
    asm volatile("" ::: "memory");
#endif
    const float* hs = &hstage[(w * NSUB + s) * SUBR * KDIM];
#pragma unroll 1
    for (int j = 0; j < SUBR; ++j) {
      const int li = w * RPW + s * SUBR + j;
      const float hv = hs[j * KDIM + lane];
      const int yr = yL[li];

      // one exp + one log give softplus(+/-h) AND sigmoid(h)
      const float t1  = __expf(-fabsf(hv));         // exp(-|h|) in (0,1]
      const float l1p = __logf(1.0f + t1);          // log1p(exp(-|h|))
      const float sp  = fmaxf(hv, 0.0f) + l1p;      // softplus(h)  = -log(1-hz)

      if (cL[li] == 0) {  // wave-uniform branch
        // uncensored: -(log Sp[y] + log hz[y]), clipped at log(EPS)
        const float s1    = waveSum32((lane < yr) ? sp : 0.0f);
        const float spn   = fmaxf(-hv, 0.0f) + l1p; // softplus(-h) = -log(hz)
        const float spn_y = __shfl(spn, yr, 32);
        acc -= fmaxf(-s1, LOGEPS) + fmaxf(-spn_y, LOGEPS);
      } else {
        // censored: 0.6*( -log Sp[y+1] - log Sp2[y+1] )
        const float s2 = waveSum32((lane <= yr) ? sp : 0.0f);
        const int pm = pmaxL[li];
        const int sm = smaxL[li];
        const float hz = ((hv >= 0.0f) ? 1.0f : t1) / (1.0f + t1);  // sigmoid(h)
        const float xv = (lane > pm) ? hz : 0.0f;   // hz * pre_mask
        const float e  = __expf(xv);                // masked entries -> exp(0)=1
        const float se = waveSum32(e);              // softmax denom (>= 32)
        const float hz2 = (lane > sm) ? (e / se) : 0.0f;  // < e/32: log1p safe
        const float lg  = (lane <= yr) ? __logf(1.0f - hz2) : 0.0f;
        const float s3  = waveSum32(lg);
        acc -= 0.6f * (fmaxf(-s2, LOGEPS) + fmaxf(s3, LOGEPS));
      }
    }
  }

  if (lane == 0) wacc[w] = acc;   // acc is wave-uniform after butterflies
  __syncthreads();
  if (i == 0) {
    float t = 0.0f;
#pragma unroll
    for (int q = 0; q < WPB; ++q) t += wacc[q];
    partial[b] = t;
  }
}

// ---------------- K4: deterministic final reduction + mean ----------------
__global__ __launch_bounds__(1024) void k4_reduce(const float* __restrict__ partial,
                                                  float* __restrict__ out) {
  __shared__ float red[1024];
  const int t = threadIdx.x;
  float v = 0.0f;
  for (int q = t; q < (int)NB; q += 1024) v += partial[q];
  red[t] = v;
  __syncthreads();
  for (int off = 512; off > 0; off >>= 1) {
    if (t < off) red[t] += red[t + off];
    __syncthreads();
  }
  if (t == 0) out[0] = red[0] * (1.0f / (float)BROWS);
}

extern "C" void kernel_launch(void* const* d_in, const int* in_sizes, int n_in,
                              void* d_out, int out_size, void* d_ws, size_t ws_size,
                              hipStream_t stream) {
  const float* h = (const float*)d_in[0];
  const int*   y = (const int*)d_in[1];
  // d_in[2] = t : unused by the reference loss
  const int*   c = (const int*)d_in[3];

  int*   bmax = (int*)d_ws;         // NB ints
  int*   fwd  = bmax + NB;          // NB ints
  int*   bwd  = fwd + NB;           // NB ints
  float* part = (float*)(bwd + NB); // NB floats
  float* out  = (float*)d_out;

  k1_blockmax<<<NB, CHUNK, 0, stream>>>(y, c, bmax);
  k2_scan<<<1, 1024, 0, stream>>>(bmax, fwd, bwd);
  k3_main<<<NB, CHUNK, 0, stream>>>(h, y, c, fwd, bwd, part);
  k4_reduce<<<1, 1024, 0, stream>>>(part, out);
}